// DynamicPillarFeatureNet_8942121910502
// MI455X (gfx1250) — compile-verified
//
#include <hip/hip_runtime.h>

typedef __attribute__((ext_vector_type(2))) float v2f;
typedef __attribute__((ext_vector_type(8))) float v8f;

// Problem constants (fixed by the reference setup)
#define PTS_B   4
#define PTS_P   200000
#define PTS_F   5
#define GRID_W  704
#define GRID_H  800
#define N_CH    32
#define PLANE   (GRID_W * GRID_H)        // 563200
#define PX0     0.0f
#define PY0     (-40.0f)
#define PZ0     (-3.0f)
#define PSIZE   0.1f
#define CZMID   2.0f                     // (1 - (-3)) / 2
#define BN_EPS  1e-3f

// ---------------------------------------------------------------------------
// Fold BN (eval mode) into the weights:  h = ReLU(feat @ Wp^T) with
// Wp[c][k] = s[c]*W[c][k] (k<11), Wp[c][11] = beta[c] - s[c]*mean[c],
// s[c] = gamma[c] * rsqrt(var[c] + eps).  Feature 11 is a constant 1.0.
// ---------------------------------------------------------------------------
__global__ void fold_bn_kernel(const float* __restrict__ W,
                               const float* __restrict__ gamma,
                               const float* __restrict__ beta,
                               const float* __restrict__ mean,
                               const float* __restrict__ var,
                               float* __restrict__ Wp) {
    int c = threadIdx.x;
    if (c < N_CH) {
        float s = gamma[c] * rsqrtf(var[c] + BN_EPS);
        #pragma unroll
        for (int k = 0; k < 11; ++k) Wp[c * 12 + k] = s * W[c * 11 + k];
        Wp[c * 12 + 11] = beta[c] - s * mean[c];
    }
}

// ---------------------------------------------------------------------------
// Zero-fill the dense output grid with 128-bit stores (dominant BW cost).
// ---------------------------------------------------------------------------
__global__ void zero_out_kernel(float4* __restrict__ out, int n4) {
    int i = blockIdx.x * blockDim.x + threadIdx.x;
    if (i < n4) out[i] = make_float4(0.f, 0.f, 0.f, 0.f);
}

// ---------------------------------------------------------------------------
// Fused point pass: features -> WMMA f32 16x16x4 MLP -> ReLU -> atomic max
// One wave = one tile of 16 points x 32 channels.
// ---------------------------------------------------------------------------
__global__ void __launch_bounds__(256)
pillar_wmma_kernel(const float* __restrict__ pts,
                   const float* __restrict__ Wp,
                   unsigned int* __restrict__ out,
                   int ntiles) {
    const int lane   = threadIdx.x & 31;
    const int wid    = blockIdx.x * (blockDim.x >> 5) + (threadIdx.x >> 5);
    const int nwaves = gridDim.x * (blockDim.x >> 5);
    const bool hiK   = lane >= 16;          // upper half-wave holds K+2/K+3 (A) / K rows 2,3 (B)
    const int  n0    = lane & 15;

    // B-matrix fragments (loop invariant): chunk c covers K = 4c..4c+3,
    // half h covers channels N = 16h..16h+15.
    v2f bm[3][2];
    #pragma unroll
    for (int c = 0; c < 3; ++c) {
        #pragma unroll
        for (int h = 0; h < 2; ++h) {
            const float* wr = Wp + (n0 + 16 * h) * 12 + 4 * c;
            bm[c][h].x = wr[hiK ? 2 : 0];
            bm[c][h].y = wr[hiK ? 3 : 1];
        }
    }

    for (int t = wid; t < ntiles; t += nwaves) {
        const int g0 = t * 16;              // first global point of tile
        const int g  = g0 + n0;             // this lane's point (both half-waves)
        const int b  = g / PTS_P;           // whole tile is in one batch (P % 16 == 0)

        const float* p = pts + (size_t)g * PTS_F;
        float x = p[0], y = p[1], z = p[2], f3 = p[3], f4 = p[4];
        float rx = x - PX0, ry = y - PY0, rz = z - PZ0;
        int ix = (int)floorf(rx / PSIZE);
        int iy = (int)floorf(ry / PSIZE);
        float ox = rx - ((float)ix + 0.5f) * PSIZE;
        float oy = ry - ((float)iy + 0.5f) * PSIZE;
        float oz = rz - CZMID;
        // 12 features: [x,y,z,f3,f4, rx,ry,rz, ox,oy,oz, 1]
        float f0 = x,  f1 = y,  f2 = z,  f3_ = f3, f4_ = f4;
        float f5 = rx, f6 = ry, f7 = rz, f8 = ox, f9 = oy, f10 = oz, f11 = 1.0f;

        bool valid = (ix >= 0) & (ix < GRID_W) & (iy >= 0) & (iy < GRID_H);
        int pbase  = valid ? (b * (N_CH * PLANE) + iy * GRID_W + ix) : -1;

        v8f acc0 = {};   // channels 0..15
        v8f acc1 = {};   // channels 16..31

        // K chunk 0: features 0..3
        {
            v2f a; a.x = hiK ? f2 : f0;  a.y = hiK ? f3_ : f1;
            acc0 = __builtin_amdgcn_wmma_f32_16x16x4_f32(false, a, false, bm[0][0], (short)0, acc0, false, false);
            acc1 = __builtin_amdgcn_wmma_f32_16x16x4_f32(false, a, false, bm[0][1], (short)0, acc1, false, false);
        }
        // K chunk 1: features 4..7
        {
            v2f a; a.x = hiK ? f6 : f4_; a.y = hiK ? f7 : f5;
            acc0 = __builtin_amdgcn_wmma_f32_16x16x4_f32(false, a, false, bm[1][0], (short)0, acc0, false, false);
            acc1 = __builtin_amdgcn_wmma_f32_16x16x4_f32(false, a, false, bm[1][1], (short)0, acc1, false, false);
        }
        // K chunk 2: features 8..11
        {
            v2f a; a.x = hiK ? f10 : f8; a.y = hiK ? f11 : f9;
            acc0 = __builtin_amdgcn_wmma_f32_16x16x4_f32(false, a, false, bm[2][0], (short)0, acc0, false, false);
            acc1 = __builtin_amdgcn_wmma_f32_16x16x4_f32(false, a, false, bm[2][1], (short)0, acc1, false, false);
        }

        // Scatter: D VGPR r / lane L holds point M = r + 8*(L>=16), channel n0 (+16).
        // ReLU folded into the >0 predicate (grid is zero-initialized, values >=0
        // so uint max == float max on the bit pattern).
        #pragma unroll
        for (int r = 0; r < 8; ++r) {
            int m  = r + (hiK ? 8 : 0);
            int pb = __shfl(pbase, m, 32);
            if (pb >= 0) {
                float v0 = acc0[r];
                float v1 = acc1[r];
                if (v0 > 0.0f)
                    atomicMax(out + pb + n0 * PLANE, __float_as_uint(v0));
                if (v1 > 0.0f)
                    atomicMax(out + pb + (n0 + 16) * PLANE, __float_as_uint(v1));
            }
        }
    }
}

// ---------------------------------------------------------------------------
extern "C" void kernel_launch(void* const* d_in, const int* in_sizes, int n_in,
                              void* d_out, int out_size, void* d_ws, size_t ws_size,
                              hipStream_t stream) {
    const float* pts   = (const float*)d_in[0];
    const float* W     = (const float*)d_in[1];
    const float* gamma = (const float*)d_in[2];
    const float* beta  = (const float*)d_in[3];
    const float* mean  = (const float*)d_in[4];
    const float* var   = (const float*)d_in[5];

    float* Wp = (float*)d_ws;               // 32 x 12 folded weights (1.5 KB)

    fold_bn_kernel<<<1, 32, 0, stream>>>(W, gamma, beta, mean, var, Wp);

    int n4 = out_size / 4;                  // out_size divisible by 4
    int zblocks = (n4 + 255) / 256;
    zero_out_kernel<<<zblocks, 256, 0, stream>>>((float4*)d_out, n4);

    int npts   = in_sizes[0] / PTS_F;       // B*P = 800000
    int ntiles = npts / 16;                 // 50000 wave-tiles
    int blocks = (ntiles + 7) / 8;          // 8 waves (wave32) per 256-thread block
    pillar_wmma_kernel<<<blocks, 256, 0, stream>>>(pts, Wp, (unsigned int*)d_out,
                                                   ntiles);
}